// SortPooling_21921513079205
// MI455X (gfx1250) — compile-verified
//
#include <hip/hip_runtime.h>
#include <math.h>

typedef __attribute__((ext_vector_type(2))) float v2f;
typedef __attribute__((ext_vector_type(8))) float v8f;

#define WAVES_PER_BLOCK 16
#define KPOOL 16
#define LPAD 17  // odd stride -> 32 lanes hit 32 distinct LDS banks

// ---------------------------------------------------------------------------
// Kernel 0: Wn = softmax(W), K=16. Trivial, single thread.
// ---------------------------------------------------------------------------
__global__ void softmax_w_kernel(const float* __restrict__ W,
                                 float* __restrict__ Wn, int k) {
  if (blockIdx.x == 0 && threadIdx.x == 0) {
    float mx = -INFINITY;
    for (int i = 0; i < k; ++i) mx = fmaxf(mx, W[i]);
    float s = 0.f;
    for (int i = 0; i < k; ++i) s += __expf(W[i] - mx);
    float inv = 1.f / s;
    for (int i = 0; i < k; ++i) Wn[i] = __expf(W[i] - mx) * inv;
  }
}

// ---------------------------------------------------------------------------
// Kernel 1: segment boundaries from sorted e_map.
// segStart[s] = first index i with e_map[i] >= s;  segStart[n] = total.
// ---------------------------------------------------------------------------
__global__ void seg_bounds_kernel(const int* __restrict__ e_map, int total,
                                  int n, int* __restrict__ segStart) {
  int i = blockIdx.x * blockDim.x + threadIdx.x;
  if (i >= total) return;
  int e = e_map[i];
  int prev = (i == 0) ? -1 : e_map[i - 1];
  for (int s = prev + 1; s <= e; ++s) segStart[s] = i;
  if (i == total - 1) {
    for (int s = e + 1; s <= n; ++s) segStart[s] = total;
  }
}

// ---------------------------------------------------------------------------
// Kernel 2: one wave32 per segment -> exact top-16, then per-block f32 WMMA
// chain computing out[16 segs] = Top(16x16) * softmax(W).
// ---------------------------------------------------------------------------
__global__ __launch_bounds__(32 * WAVES_PER_BLOCK, 1)
void sortpool_topk_kernel(const float* __restrict__ Y,
                          const int* __restrict__ segStart,
                          const float* __restrict__ Wn,
                          float* __restrict__ out, int n) {
  __shared__ float lists[WAVES_PER_BLOCK][32][LPAD];  // padded: no bank conflicts
  __shared__ float T[WAVES_PER_BLOCK][LPAD];          // Top tile (padded rows)

  const int lane = threadIdx.x & 31;
  const int wave = threadIdx.x >> 5;
  const int seg = blockIdx.x * WAVES_PER_BLOCK + wave;

  // --- per-lane register top-16 (sorted descending) via compare-insert chain
  float t[KPOOL];
#pragma unroll
  for (int j = 0; j < KPOOL; ++j) t[j] = -INFINITY;

  if (seg < n) {
    const int start = segStart[seg];
    const int end = segStart[seg + 1];
    for (int i = start + lane; i < end; i += 32) {  // coalesced 128B per wave
      float v = Y[i];
#pragma unroll
      for (int j = 0; j < KPOOL; ++j) {
        float old = t[j];
        bool gt = v > old;
        t[j] = gt ? v : old;
        v = gt ? old : v;
      }
    }
  }
#pragma unroll
  for (int j = 0; j < KPOOL; ++j) lists[wave][lane][j] = t[j];

  // --- merge: 16 rounds of wave-wide max over per-lane sorted-list heads.
  // Extracts the exact top-16 multiset (one instance removed per round,
  // matching the reference's first-occurrence masking under ties).
  int ptr = 0;
  float keep = 0.f;
  for (int r = 0; r < KPOOL; ++r) {
    float cand = (ptr < KPOOL) ? lists[wave][lane][ptr] : -INFINITY;
    float m = cand;
#pragma unroll
    for (int off = 16; off >= 1; off >>= 1)
      m = fmaxf(m, __shfl_xor(m, off, 32));
    unsigned long long ball = __ballot(cand == m);
    int winner = __ffsll((long long)ball) - 1;
    if (lane == winner) ptr++;
    if (lane == r) keep = isinf(m) ? 0.f : m;  // -inf padding -> 0
  }
  if (lane < KPOOL) T[wave][lane] = keep;

  __syncthreads();

  // --- wave 0: D = Top(16x16) x B, B = Wn replicated across all 16 columns,
  // as a chain of four exact-f32 V_WMMA_F32_16X16X4_F32 ops (C -> D).
  if (threadIdx.x < 32) {
    const int row = lane & 15;       // M index (segment-in-block)
    const int hi = (lane >> 4) & 1;  // lanes 16-31 carry K+2/K+3 slots
    v8f d = {0.f, 0.f, 0.f, 0.f, 0.f, 0.f, 0.f, 0.f};
#pragma unroll
    for (int j = 0; j < 4; ++j) {
      const int kbase = 4 * j + 2 * hi;
      v2f a, b;
      a.x = T[row][kbase];      // A (16x4 f32), ISA 32-bit A layout
      a.y = T[row][kbase + 1];
      b.x = Wn[kbase];          // B (4x16 f32): row K replicated over columns
      b.y = Wn[kbase + 1];
      d = __builtin_amdgcn_wmma_f32_16x16x4_f32(false, a, false, b, (short)0,
                                                d, false, false);
    }
    // C/D layout: lane 0 holds (M=0..7, N=0); lane 16 holds (M=8..15, N=0).
    const int base = blockIdx.x * WAVES_PER_BLOCK;
    if (base + WAVES_PER_BLOCK <= n) {
      // fast path: two b128 stores (out + 64B-aligned offsets)
      if (lane == 0) {
        float4 lo = make_float4(d[0], d[1], d[2], d[3]);
        float4 hi4 = make_float4(d[4], d[5], d[6], d[7]);
        *(float4*)(out + base + 0) = lo;
        *(float4*)(out + base + 4) = hi4;
      } else if (lane == 16) {
        float4 lo = make_float4(d[0], d[1], d[2], d[3]);
        float4 hi4 = make_float4(d[4], d[5], d[6], d[7]);
        *(float4*)(out + base + 8) = lo;
        *(float4*)(out + base + 12) = hi4;
      }
    } else {
      // rare ragged tail (n not a multiple of 16)
      if (lane == 0) {
#pragma unroll
        for (int m = 0; m < 8; ++m)
          if (base + m < n) out[base + m] = d[m];
      } else if (lane == 16) {
#pragma unroll
        for (int m = 0; m < 8; ++m)
          if (base + 8 + m < n) out[base + 8 + m] = d[m];
      }
    }
  }
}

// ---------------------------------------------------------------------------
extern "C" void kernel_launch(void* const* d_in, const int* in_sizes, int n_in,
                              void* d_out, int out_size, void* d_ws,
                              size_t ws_size, hipStream_t stream) {
  const float* Y = (const float*)d_in[0];
  const float* W = (const float*)d_in[1];
  const int* e_map = (const int*)d_in[2];
  (void)d_in[3];  // v_count: only its length is used

  const int total = in_sizes[0];  // 16,777,216
  const int k = in_sizes[1];      // 16
  const int n = in_sizes[3];      // 65,536
  float* out = (float*)d_out;

  // workspace layout: segStart[n+1] ints, then Wn[16] floats (256B aligned)
  int* segStart = (int*)d_ws;
  size_t off = (((size_t)(n + 1) * sizeof(int)) + 255) & ~(size_t)255;
  float* Wn = (float*)((char*)d_ws + off);

  softmax_w_kernel<<<1, 32, 0, stream>>>(W, Wn, k);

  {
    const int threads = 256;
    const int blocks = (total + threads - 1) / threads;
    seg_bounds_kernel<<<blocks, threads, 0, stream>>>(e_map, total, n,
                                                      segStart);
  }

  {
    const int threads = 32 * WAVES_PER_BLOCK;  // 512 = 16 waves
    const int blocks = (n + WAVES_PER_BLOCK - 1) / WAVES_PER_BLOCK;
    sortpool_topk_kernel<<<blocks, threads, 0, stream>>>(Y, segStart, Wn, out,
                                                         n);
  }
}